// HeightSumCompression_72971494359429
// MI455X (gfx1250) — compile-verified
//
#include <hip/hip_runtime.h>

// Problem constants (from reference)
#define NPTS 1000000
#define CCH  128
#define B_   8
#define Y_   468
#define X_   468
#define KTOT (B_ * Y_ * X_)          // 1,752,192 dense keys
#define TILE 256                      // elements scanned per wave via WMMA
#define WPB  8                        // waves per block
#define BLK_ELEMS (TILE * WPB)        // 2048 elements per block
#define NB0  ((KTOT + BLK_ELEMS - 1) / BLK_ELEMS)   // 856 level-0 blocks

typedef float v2f __attribute__((ext_vector_type(2)));
typedef float v8f __attribute__((ext_vector_type(8)));

// ---------------- presence / keys ----------------

__global__ void zero_presence_kernel(int* __restrict__ presence) {
    int k = blockIdx.x * blockDim.x + threadIdx.x;
    if (k < KTOT) presence[k] = 0;
}

__global__ void mark_presence_kernel(const int* __restrict__ idx,
                                     int* __restrict__ presence,
                                     int* __restrict__ keys) {
    int i = blockIdx.x * blockDim.x + threadIdx.x;
    if (i >= NPTS) return;
    int b = idx[i * 4 + 0];
    int y = idx[i * 4 + 2];
    int x = idx[i * 4 + 3];
    int k = b * (Y_ * X_) + y * X_ + x;
    keys[i] = k;
    presence[k] = 1;   // racy stores of identical value: benign
}

// ---------------- level-0 scan: WMMA tensor-scan ----------------
// Each wave scans a 256-element tile as a 16x16 matrix M (row-major,
// element e = M[e/16][e%16]).  Row-inclusive scans P = M x U with U the
// 16x16 upper-triangular ones matrix, computed as four chained
// V_WMMA_F32_16X16X4_F32 (K=4 slices accumulating through C).
// VGPR layouts per CDNA5 ISA 7.12.2:
//   A 16x4  f32: VGPR0 = {lanes0-15: K=0, lanes16-31: K=2}, VGPR1 = {K=1, K=3}
//   B 4x16  f32: VGPR0 = {lanes0-15: row K=0, lanes16-31: row K=2}, VGPR1 = {K=1, K=3}
//   D 16x16 f32: VGPR v = {lanes0-15: M=v, lanes16-31: M=v+8}, N = lane%16
__global__ __launch_bounds__(256)
void scan_tiles_wmma_kernel(const int* __restrict__ presence,
                            int* __restrict__ rank,
                            int* __restrict__ blockSums) {
    __shared__ float tiles[WPB][TILE];
    __shared__ float rsArr[WPB][16];    // per-row sums of each tile
    __shared__ float waveTot[WPB];

    const int w    = threadIdx.x >> 5;
    const int lane = threadIdx.x & 31;
    const int tileBase = blockIdx.x * BLK_ELEMS + w * TILE;

    // Stage presence bits (as f32 0/1) into LDS, coalesced.
    for (int j = 0; j < 8; ++j) {
        int e = lane + j * 32;
        int g = tileBase + e;
        tiles[w][e] = (g < KTOT && presence[g]) ? 1.0f : 0.0f;
    }
    __syncthreads();

    const int r0   = lane & 15;                 // N (and A-row) index
    const int kOff = (lane < 16) ? 0 : 2;       // K half selector
    const int rowBase = (lane < 16) ? 0 : 8;    // D-layout row half

    v8f acc = {0.f, 0.f, 0.f, 0.f, 0.f, 0.f, 0.f, 0.f};
    #pragma unroll
    for (int kb = 0; kb < 4; ++kb) {
        v2f a, b;
        // A slice = M[:, 4kb .. 4kb+3]
        a.x = tiles[w][r0 * 16 + 4 * kb + kOff + 0];
        a.y = tiles[w][r0 * 16 + 4 * kb + kOff + 1];
        // B slice = U[4kb .. 4kb+3, :],  U[k][n] = (k <= n)
        b.x = ((4 * kb + kOff + 0) <= r0) ? 1.0f : 0.0f;
        b.y = ((4 * kb + kOff + 1) <= r0) ? 1.0f : 0.0f;
        acc = __builtin_amdgcn_wmma_f32_16x16x4_f32(
            /*neg_a=*/false, a, /*neg_b=*/false, b,
            /*c_mod=*/(short)0, acc, /*reuse_a=*/false, /*reuse_b=*/false);
    }
    // acc[v] = P[rowBase+v][r0] = inclusive scan within row rowBase+v.

    // Row sums live in column 15 (lanes 15 and 31).
    if (r0 == 15) {
        #pragma unroll
        for (int v = 0; v < 8; ++v) rsArr[w][rowBase + v] = acc[v];
    }
    __syncthreads();

    // Per-lane cross-row offsets (register-resident, constant indexing).
    float base = 0.0f;
    if (lane >= 16) {
        #pragma unroll
        for (int r = 0; r < 8; ++r) base += rsArr[w][r];
    }
    float offs[8];
    float run = base;
    #pragma unroll
    for (int v = 0; v < 8; ++v) {
        offs[v] = run;
        run += rsArr[w][rowBase + v];
    }
    // lane 31: run == full tile total
    if (lane == 31) waveTot[w] = run;
    __syncthreads();

    float tileOff = 0.0f, blockTot = 0.0f;
    #pragma unroll
    for (int ww = 0; ww < WPB; ++ww) {
        float t = waveTot[ww];
        if (ww < w) tileOff += t;
        blockTot += t;
    }

    // Exclusive scan value = inclusive - own element + cross-row + cross-tile.
    #pragma unroll
    for (int v = 0; v < 8; ++v) {
        int e = (rowBase + v) * 16 + r0;
        int g = tileBase + e;
        if (g < KTOT) {
            float excl = acc[v] + offs[v] - tiles[w][e] + tileOff;
            rank[g] = (int)(excl + 0.5f);   // exact small integers in f32
        }
    }
    if (threadIdx.x == 0) blockSums[blockIdx.x] = (int)(blockTot + 0.5f);
}

// ---------------- level-1 scan: 856 block sums, one block ----------------

__global__ __launch_bounds__(1024)
void scan_blocksums_kernel(int* __restrict__ blockSums, int* __restrict__ total) {
    __shared__ int s[1024];
    int t = threadIdx.x;
    int orig = (t < NB0) ? blockSums[t] : 0;
    s[t] = orig;
    __syncthreads();
    for (int d = 1; d < 1024; d <<= 1) {
        int add = (t >= d) ? s[t - d] : 0;
        __syncthreads();
        s[t] += add;
        __syncthreads();
    }
    int excl = (t == 0) ? 0 : s[t - 1];
    if (t < NB0) blockSums[t] = excl;
    if (t == NB0 - 1) *total = s[t];     // number of unique keys
}

__global__ void add_offsets_kernel(int* __restrict__ rank,
                                   const int* __restrict__ blockSums) {
    int k = blockIdx.x * blockDim.x + threadIdx.x;
    if (k < KTOT) rank[k] += blockSums[k / BLK_ELEMS];
}

// ---------------- unique-index outputs ----------------

__global__ void write_indices_kernel(const int* __restrict__ presence,
                                     const int* __restrict__ rank,
                                     float* __restrict__ out) {
    int k = blockIdx.x * blockDim.x + threadIdx.x;
    if (k >= KTOT || !presence[k]) return;
    int r   = rank[k];
    int bb  = k / (Y_ * X_);
    int rem = k % (Y_ * X_);
    float* p = out + (size_t)NPTS * CCH + (size_t)r * 3;
    p[0] = (float)bb;
    p[1] = (float)(rem / X_);
    p[2] = (float)(rem % X_);
}

// Padded rows: decode of key -1 with Python floor/mod -> (-1, 467, 467)
__global__ void fill_padded_kernel(const int* __restrict__ total,
                                   float* __restrict__ out) {
    int r = blockIdx.x * blockDim.x + threadIdx.x;
    if (r >= NPTS) return;
    if (r >= *total) {
        float* p = out + (size_t)NPTS * CCH + (size_t)r * 3;
        p[0] = -1.0f; p[1] = 467.0f; p[2] = 467.0f;
    }
}

// ---------------- feature output ----------------

__global__ void zero_features_kernel(float4* __restrict__ out) {
    long long i = (long long)blockIdx.x * blockDim.x + threadIdx.x;
    if (i < (long long)NPTS * CCH / 4) {
        out[i] = make_float4(0.f, 0.f, 0.f, 0.f);
    }
}

// One wave per point: coalesced 512B row read, 4 f32 atomics per lane.
__global__ __launch_bounds__(256)
void scatter_add_kernel(const float* __restrict__ features,
                        const int* __restrict__ keys,
                        const int* __restrict__ rank,
                        float* __restrict__ out) {
    int gw   = (blockIdx.x * blockDim.x + threadIdx.x) >> 5;
    int lane = threadIdx.x & 31;
    if (gw >= NPTS) return;
    int r = rank[keys[gw]];
    const float4 v = ((const float4*)(features + (size_t)gw * CCH))[lane];
    float* dst = out + (size_t)r * CCH + lane * 4;
    atomicAdd(dst + 0, v.x);
    atomicAdd(dst + 1, v.y);
    atomicAdd(dst + 2, v.z);
    atomicAdd(dst + 3, v.w);
}

// ---------------- launcher ----------------

extern "C" void kernel_launch(void* const* d_in, const int* in_sizes, int n_in,
                              void* d_out, int out_size, void* d_ws, size_t ws_size,
                              hipStream_t stream) {
    const float* features = (const float*)d_in[0];
    const int*   indices  = (const int*)d_in[1];
    float*       out      = (float*)d_out;

    // Workspace layout (ints): presence[KTOT] rank[KTOT] keys[NPTS] blockSums[NB0] total[1]
    int* presence  = (int*)d_ws;
    int* rank      = presence + KTOT;
    int* keys      = rank + KTOT;
    int* blockSums = keys + NPTS;
    int* total     = blockSums + NB0;

    const int T = 256;
    const int gK = (KTOT + T - 1) / T;       // 6845
    const int gN = (NPTS + T - 1) / T;       // 3907
    const int gF = (NPTS * (CCH / 4) + T - 1) / T;   // zero features (float4)
    const int gS = (NPTS * 32 + T - 1) / T;  // one wave (32 thr) per point

    zero_presence_kernel<<<gK, T, 0, stream>>>(presence);
    mark_presence_kernel<<<gN, T, 0, stream>>>(indices, presence, keys);
    scan_tiles_wmma_kernel<<<NB0, T, 0, stream>>>(presence, rank, blockSums);
    scan_blocksums_kernel<<<1, 1024, 0, stream>>>(blockSums, total);
    add_offsets_kernel<<<gK, T, 0, stream>>>(rank, blockSums);
    write_indices_kernel<<<gK, T, 0, stream>>>(presence, rank, out);
    fill_padded_kernel<<<gN, T, 0, stream>>>(total, out);
    zero_features_kernel<<<gF, T, 0, stream>>>((float4*)out);
    scatter_add_kernel<<<gS, T, 0, stream>>>(features, keys, rank, out);
}